// LinearFunctional2d_78374563217834
// MI455X (gfx1250) — compile-verified
//
#include <hip/hip_runtime.h>
#include <math.h>

// ---------------------------------------------------------------------------
// LinearFunctional2d on MI455X (gfx1250)
//
// S[b,c] = sum_{h,w} x[b,c,h,w] * G[c,h,w]   (G complex, b-independent)
// G_c(256h x 512n) = A(256h x 128k) @ B_c(128k x 512n)   via V_WMMA_F32_16X16X4_F32
//   A = [cos(2*pi*f_j*h/256) | sin(...)],  f_j = j (j<=32), j-64 (j>32)
//   B packs weight-derived mode sums P1..P4 (Hermitian folding, corner
//   real-forcing corrections, 1/65536 forward-FFT norm).
// A and B are stored K-pair interleaved ([k/2][col][k%2]) so each WMMA
// fragment is one aligned b64 load per lane.
// Final stage: streaming fp32 dot product; x via non-temporal B128 loads
// (512 MB read exactly once from HBM), G served from L2 (33.5 MB < 192 MB),
// with 8 batch images per block to amortize G's L2 traffic 8x.
// ---------------------------------------------------------------------------

typedef float v2f __attribute__((ext_vector_type(2)));
typedef float v4f __attribute__((ext_vector_type(4)));
typedef float v8f __attribute__((ext_vector_type(8)));

#define AT_OFF   0u                         // 128*256 floats  (A packed)
#define BM_OFF   32768u                     // 64*128*512 floats (B packed)
#define GR_OFF   (32768u + 4194304u)        // 64*65536 floats (Re G planes)
#define GI_OFF   (GR_OFF + 4194304u)        // 64*65536 floats (Im G planes)
#define PT_OFF   (GI_OFF + 4194304u)        // 2048*16*2 floats (partials)
#define NSLICES  16
#define BGRP     8                          // batch images per reduce block
#define TWO_PI_256 0.0245436926f            // 2*pi/256

// ---- Kernel 1a: packed A basis, A[h][k'] with k'<64 -> cos, >=64 -> sin ----
// packed addr = (k'>>1)*512 + h*2 + (k'&1)
__global__ void k_build_AT(float* __restrict__ ws) {
    int kp = blockIdx.x;           // k' = 0..127
    int h  = threadIdx.x;          // 0..255
    int jj = kp & 63;
    int f  = (jj <= 32) ? jj : jj - 64;
    int n  = (((f * h) % 256) + 256) & 255;
    float s, c;
    sincosf((float)n * TWO_PI_256, &s, &c);
    ws[AT_OFF + (kp >> 1) * 512 + h * 2 + (kp & 1)] = (kp < 64) ? c : s;
}

// ---- Kernel 1b: packed B_c from weights (complex64, interleaved) -----------
// packed addr (per c) = (k'>>1)*1024 + n*2 + (k'&1),  n in 0..511
__global__ void k_build_B(const float* __restrict__ W, float* __restrict__ ws) {
    __shared__ float tc[256], ts[256];
    int t = threadIdx.x;
    {
        float s, c;
        sincosf((float)t * TWO_PI_256, &s, &c);
        tc[t] = c; ts[t] = s;
    }
    __syncthreads();

    int c = blockIdx.x >> 6;       // channel 0..63
    int j = blockIdx.x & 63;       // mode row 0..63
    int w = t;                     // 0..255
    const float* Wrow = W + (size_t)(c * 64 + j) * 33 * 2;

    float p1 = 0.f, p2 = 0.f, p3 = 0.f, p4 = 0.f;
    for (int k = 0; k <= 32; ++k) {
        float wr = Wrow[2 * k];
        float wi = Wrow[2 * k + 1];
        bool edge = (k == 0) | (k == 32);
        if (edge & ((j == 0) | (j == 32))) wi = 0.f;     // corner real-forcing (W side)
        int n = (k * w) & 255;
        float ck = tc[n], sk = ts[n];
        float g = edge ? 1.f : 2.f;                       // Hermitian fold weight
        p1 += g * (ck * wr - sk * wi);
        p2 += g * (ck * wi + sk * wr);
        if (edge) {                                       // k in {0,32}: complex-kept cols
            p3 += ck * wi + sk * wr;
            p4 += ck * wr - sk * wi;
        }
    }
    // Corner corrections for X-side real-forcing (drop i*sin(theta_total)*Wr)
    int n32 = (32 * w) & 255;
    float c32 = tc[n32], s32 = ts[n32];
    const float* Wb = W + (size_t)c * 64 * 33 * 2;
    if (j == 0) {
        p3 -= s32 * Wb[(0 * 33 + 32) * 2];
    }
    if (j == 32) {
        float wr_32_0  = Wb[(32 * 33 + 0) * 2];
        float wr_32_32 = Wb[(32 * 33 + 32) * 2];
        p4 -= wr_32_0 + c32 * wr_32_32;
        p3 -= s32 * wr_32_32;
    }
    const float invN = 1.0f / 65536.0f;                   // rfft2 norm='forward'
    float* Bm = ws + BM_OFF + (size_t)c * 128 * 512;
    int j2 = 64 + j;
    // row j  -> P1 (Re cols), P3 (Im cols); row 64+j -> -P2, P4
    Bm[(j  >> 1) * 1024 + w * 2 + (j  & 1)]         =  p1 * invN;
    Bm[(j2 >> 1) * 1024 + w * 2 + (j2 & 1)]         = -p2 * invN;
    Bm[(j  >> 1) * 1024 + (256 + w) * 2 + (j  & 1)] =  p3 * invN;
    Bm[(j2 >> 1) * 1024 + (256 + w) * 2 + (j2 & 1)] =  p4 * invN;
}

// ---- Kernel 2: 64x (256x512 = 256x128 @ 128x512) GEMM via f32 WMMA ---------
// 8 waves/block, one 16x16 tile per wave, K in steps of 4; b64 fragment loads.
__global__ void __launch_bounds__(256) k_gemm_G(float* __restrict__ ws) {
    int lane = threadIdx.x & 31;
    int gw   = blockIdx.x * 8 + (threadIdx.x >> 5);   // 0..32767
    int c    = gw >> 9;
    int rem  = gw & 511;
    int tm   = rem >> 5;        // 0..15  (h tile)
    int tn   = rem & 31;        // 0..31  (n tile; <16 -> Re, >=16 -> Im)

    const float* AP = ws + AT_OFF;
    const float* BP = ws + BM_OFF + (size_t)c * 65536;

    int lrow  = lane & 15;
    int khalf = (lane >> 4) * 2;          // ISA A-layout: lanes 16-31 hold K=2,3
    int hrow  = tm * 16 + lrow;
    int col   = tn * 16 + lrow;

    v8f acc = {};
    #pragma unroll 4
    for (int kk = 0; kk < 32; ++kk) {
        int kp = kk * 2 + (khalf >> 1);   // K-pair index (k0 = 2*kp)
        v2f a = *(const v2f*)(AP + kp * 512  + hrow * 2);  // A[hrow][k0..k0+1]
        v2f b = *(const v2f*)(BP + kp * 1024 + col  * 2);  // B[k0..k0+1][col]
        acc = __builtin_amdgcn_wmma_f32_16x16x4_f32(
                  false, a, false, b, (short)0, acc, false, false);
    }

    float* plane = (tn < 16) ? (ws + GR_OFF) : (ws + GI_OFF);
    int ncol  = (tn < 16) ? col : (col - 256);
    int mbase = tm * 16 + (lane >> 4) * 8;           // ISA C/D layout: M=v / v+8
    #pragma unroll
    for (int v = 0; v < 8; ++v) {
        plane[(size_t)c * 65536 + (mbase + v) * 256 + ncol] = acc[v];
    }
}

// ---- Kernel 3: streaming weighted reduction over x (HBM-bound stage) -------
// One block = one (c, slice) G chunk x 8 batch images -> G L2 traffic / 8.
__global__ void __launch_bounds__(256) k_reduce(const float* __restrict__ x,
                                                float* __restrict__ ws) {
    int t    = threadIdx.x;
    int lane = t & 31;
    int wv   = t >> 5;
    int bi   = blockIdx.x;                 // 0..4095
    int bg   = bi >> 10;                   // batch group 0..3 (b = bg*8+u)
    int c    = (bi >> 4) & 63;
    int s    = bi & (NSLICES - 1);

    const v4f* gr4 = (const v4f*)(ws + GR_OFF + (size_t)c * 65536 + s * 4096);
    const v4f* gi4 = (const v4f*)(ws + GI_OFF + (size_t)c * 65536 + s * 4096);
    const v4f* x4[BGRP];
    #pragma unroll
    for (int u = 0; u < BGRP; ++u) {
        int bc = (bg * BGRP + u) * 64 + c;
        x4[u] = (const v4f*)(x + (size_t)bc * 65536 + s * 4096);
    }

    float sr[BGRP] = {}, si[BGRP] = {};
    #pragma unroll
    for (int i = 0; i < 4; ++i) {
        int idx = i * 256 + t;
        v4f gr = gr4[idx];                               // G stays hot in L2
        v4f gi = gi4[idx];
        #pragma unroll
        for (int u = 0; u < BGRP; ++u) {
            v4f xv = __builtin_nontemporal_load(&x4[u][idx]);  // x read once
            sr[u] += xv.x * gr.x + xv.y * gr.y + xv.z * gr.z + xv.w * gr.w;
            si[u] += xv.x * gi.x + xv.y * gi.y + xv.z * gi.z + xv.w * gi.w;
        }
    }

    // wave32 butterfly reduce, then cross-wave combine in LDS
    #pragma unroll
    for (int u = 0; u < BGRP; ++u) {
        #pragma unroll
        for (int off = 16; off > 0; off >>= 1) {
            sr[u] += __shfl_xor(sr[u], off, 32);
            si[u] += __shfl_xor(si[u], off, 32);
        }
    }
    __shared__ float red[8][BGRP * 2];
    if (lane == 0) {
        #pragma unroll
        for (int u = 0; u < BGRP; ++u) {
            red[wv][u * 2]     = sr[u];
            red[wv][u * 2 + 1] = si[u];
        }
    }
    __syncthreads();
    if (t < BGRP * 2) {
        float v = 0.f;
        #pragma unroll
        for (int w8 = 0; w8 < 8; ++w8) v += red[w8][t];
        int u    = t >> 1;
        int comp = t & 1;
        int bc   = (bg * BGRP + u) * 64 + c;
        ws[PT_OFF + (bc * NSLICES + s) * 2 + comp] = v;
    }
}

// ---- Kernel 4: combine partials -> complex64 output (overwrites d_out) -----
__global__ void k_combine(const float* __restrict__ ws, float* __restrict__ out) {
    int bc = blockIdx.x * 256 + threadIdx.x;
    if (bc >= 2048) return;
    const float* part = ws + PT_OFF;
    float sr = 0.f, si = 0.f;
    #pragma unroll
    for (int s = 0; s < NSLICES; ++s) {
        sr += part[(bc * NSLICES + s) * 2];
        si += part[(bc * NSLICES + s) * 2 + 1];
    }
    out[bc * 2]     = sr;
    out[bc * 2 + 1] = si;
}

extern "C" void kernel_launch(void* const* d_in, const int* in_sizes, int n_in,
                              void* d_out, int out_size, void* d_ws, size_t ws_size,
                              hipStream_t stream) {
    const float* x = (const float*)d_in[0];   // (32,64,256,256) fp32
    const float* w = (const float*)d_in[1];   // (64,64,33) complex64 interleaved
    float* ws  = (float*)d_ws;
    float* out = (float*)d_out;               // (32,64) complex64 interleaved

    k_build_AT<<<128,  256, 0, stream>>>(ws);
    k_build_B <<<4096, 256, 0, stream>>>(w, ws);
    k_gemm_G  <<<4096, 256, 0, stream>>>(ws);
    k_reduce  <<<4096, 256, 0, stream>>>(x, ws);
    k_combine <<<8,    256, 0, stream>>>(ws, out);
}